// MessageGraphConvolution_77601469104330
// MI455X (gfx1250) — compile-verified
//
#include <hip/hip_runtime.h>

// MessageGraphConvolution for MI455X (gfx1250, wave32).
//   out = (segment_sum(x[dst], src) / deg) @ W^T + x @ B^T
// Round 4: replace the 80M-f32-atomic scatter with a CSR gather:
//   count -> prefix scan -> bin edges by source -> one wave per node
//   accumulates neighbor rows in registers (coalesced b128 loads, no float
//   atomics) and writes the normalized agg row once.
// GEMM phase unchanged: pre-swizzled W/B operand images + V_WMMA_F32_16X16X4_F32.

typedef __attribute__((ext_vector_type(2))) float v2f;
typedef __attribute__((ext_vector_type(4))) float v4f;
typedef __attribute__((ext_vector_type(8))) float v8f;

#define NNODES 100000
#define NEDGES 625000
#define FDIM   128
#define NBLK   ((NNODES + 255) / 256)   // 391 scan blocks

// ---------------- zero int scratch (cnt + cursor) ----------------
__global__ __launch_bounds__(256) void gcn_zero_kernel(int* __restrict__ cnt) {
  const int i = blockIdx.x * blockDim.x + threadIdx.x;
  if (i < 2 * NNODES) cnt[i] = 0;   // cnt[N] then cursor[N], contiguous
}

// ---------------- swizzle W and B into WMMA operand images ----------
// img[t][ks2][lane][4] floats; lane r=l&15 -> o=t*16+r, kh=l>>4, k0=ks2*8:
// {W[o][k0+2kh], W[o][k0+2kh+1], W[o][k0+4+2kh], W[o][k0+4+2kh+1]}
__global__ __launch_bounds__(256) void gcn_swizzle_kernel(
    const float* __restrict__ W, const float* __restrict__ Bm,
    float* __restrict__ wswz, float* __restrict__ bswz) {
  const int i = blockIdx.x * blockDim.x + threadIdx.x;  // 0..8191
  if (i >= 2 * 4096) return;
  const float* src = (i < 4096) ? W : Bm;
  float* dst = (i < 4096) ? wswz : bswz;
  const int j   = i & 4095;
  const int t   = j >> 9;
  const int ks2 = (j >> 5) & 15;
  const int l   = j & 31;
  const int o   = t * 16 + (l & 15);
  const int k0  = ks2 * 8 + 2 * (l >> 4);
  float4 v;
  v.x = src[(size_t)o * FDIM + k0];
  v.y = src[(size_t)o * FDIM + k0 + 1];
  v.z = src[(size_t)o * FDIM + k0 + 4];
  v.w = src[(size_t)o * FDIM + k0 + 5];
  ((float4*)dst)[j] = v;
}

// ---------------- CSR build: count, scan, bin ----------------
__global__ __launch_bounds__(256) void gcn_count_kernel(const int* __restrict__ ei,
                                                        int* __restrict__ cnt) {
  const int e = blockIdx.x * blockDim.x + threadIdx.x;
  if (e < NEDGES) atomicAdd(&cnt[ei[e]], 1);
}

__global__ __launch_bounds__(256) void gcn_scan_reduce_kernel(
    const int* __restrict__ cnt, int* __restrict__ blksum) {
  __shared__ int sdata[256];
  const int i = blockIdx.x * 256 + threadIdx.x;
  sdata[threadIdx.x] = (i < NNODES) ? cnt[i] : 0;
  __syncthreads();
  for (int s = 128; s > 0; s >>= 1) {
    if (threadIdx.x < (unsigned)s) sdata[threadIdx.x] += sdata[threadIdx.x + s];
    __syncthreads();
  }
  if (threadIdx.x == 0) blksum[blockIdx.x] = sdata[0];
}

__global__ void gcn_scan_top_kernel(const int* __restrict__ blksum,
                                    int* __restrict__ blkoff,
                                    int* __restrict__ offs) {
  if (threadIdx.x == 0 && blockIdx.x == 0) {
    int run = 0;
    for (int b = 0; b < NBLK; ++b) { blkoff[b] = run; run += blksum[b]; }
    offs[NNODES] = run;   // == NEDGES
  }
}

__global__ __launch_bounds__(256) void gcn_scan_apply_kernel(
    const int* __restrict__ cnt, const int* __restrict__ blkoff,
    int* __restrict__ offs) {
  __shared__ int sdata[256];
  const int i = blockIdx.x * 256 + threadIdx.x;
  const int v = (i < NNODES) ? cnt[i] : 0;
  sdata[threadIdx.x] = v;
  __syncthreads();
  for (int s = 1; s < 256; s <<= 1) {           // inclusive Hillis-Steele
    const int t = (threadIdx.x >= (unsigned)s) ? sdata[threadIdx.x - s] : 0;
    __syncthreads();
    sdata[threadIdx.x] += t;
    __syncthreads();
  }
  if (i < NNODES) offs[i] = blkoff[blockIdx.x] + sdata[threadIdx.x] - v;  // exclusive
}

__global__ __launch_bounds__(256) void gcn_bin_kernel(
    const int* __restrict__ ei, const int* __restrict__ offs,
    int* __restrict__ cur, int* __restrict__ bind) {
  const int e = blockIdx.x * blockDim.x + threadIdx.x;
  if (e < NEDGES) {
    const int s = ei[e];
    const int p = offs[s] + atomicAdd(&cur[s], 1);
    bind[p] = ei[NEDGES + e];
  }
}

// ---------------- gather-aggregate: one wave per node, no float atomics ----
// Lane L accumulates floats [4L, 4L+4) of the 128-wide row; writes the
// degree-normalized agg row exactly once.
__global__ __launch_bounds__(256) void gcn_aggregate_kernel(
    const float* __restrict__ x, const int* __restrict__ offs,
    const int* __restrict__ bind, float* __restrict__ agg) {
  const int n = blockIdx.x * (blockDim.x >> 5) + (threadIdx.x >> 5);
  if (n >= NNODES) return;                 // wave-uniform exit
  const int lane = threadIdx.x & 31;
  const int beg = offs[n];
  const int end = offs[n + 1];
  float4 acc = {0.0f, 0.0f, 0.0f, 0.0f};
  for (int e = beg; e < end; ++e) {
    const int d = __builtin_amdgcn_readfirstlane(bind[e]);  // wave-uniform dst
    const float4 v = ((const float4*)(x + (size_t)d * FDIM))[lane];
    acc.x += v.x; acc.y += v.y; acc.z += v.z; acc.w += v.w;
  }
  const float invd = (end > beg) ? (1.0f / (float)(end - beg)) : 1.0f;
  acc.x *= invd; acc.y *= invd; acc.z *= invd; acc.w *= invd;
  ((float4*)(agg + (size_t)n * FDIM))[lane] = acc;
}

// ---------------- dual GEMM via f32 WMMA ----------------
// Each wave: 32 nodes x 128 outputs = 16 tiles of 16x16 (acc in 128 VGPRs).
// A operand (16x4 f32): lanes 0-15 hold K={k0,k0+1} of row M=lane,
//                       lanes 16-31 hold K={k0+2,k0+3} of row M=lane-16.
// B operand comes pre-swizzled: one b128 per (t, ks2) holds 2 k-steps.
__global__ __launch_bounds__(256) void gcn_gemm_kernel(
    const float* __restrict__ x,
    const float* __restrict__ agg,     // already degree-normalized
    const float* __restrict__ wswz,
    const float* __restrict__ bswz,
    float* __restrict__ out) {
  const int wid = blockIdx.x * (blockDim.x >> 5) + (threadIdx.x >> 5);
  if (wid >= NNODES / 32) return;          // wave-uniform: EXEC stays all-ones
  const int lane = threadIdx.x & 31;
  const int row  = lane & 15;
  const int kh   = lane >> 4;
  const int m0   = wid * 32;

  const int n0 = m0 + row;
  const int n1 = m0 + 16 + row;
  const size_t koff = (size_t)(kh * 2);
  const float* a0p = agg + (size_t)n0 * FDIM + koff;
  const float* a1p = agg + (size_t)n1 * FDIM + koff;
  const float* x0p = x   + (size_t)n0 * FDIM + koff;
  const float* x1p = x   + (size_t)n1 * FDIM + koff;

  v8f acc0[8] = {};
  v8f acc1[8] = {};

  for (int ks2 = 0; ks2 < 16; ++ks2) {
    const int k0 = ks2 * 8;
    const v2f a0l = *(const v2f*)(a0p + k0);
    const v2f a0h = *(const v2f*)(a0p + k0 + 4);
    const v2f a1l = *(const v2f*)(a1p + k0);
    const v2f a1h = *(const v2f*)(a1p + k0 + 4);
    const v2f x0l = *(const v2f*)(x0p + k0);
    const v2f x0h = *(const v2f*)(x0p + k0 + 4);
    const v2f x1l = *(const v2f*)(x1p + k0);
    const v2f x1h = *(const v2f*)(x1p + k0 + 4);

    const float* wp = wswz + (size_t)ks2 * 128 + (size_t)lane * 4;
    const float* bp = bswz + (size_t)ks2 * 128 + (size_t)lane * 4;
#pragma unroll
    for (int t = 0; t < 8; ++t) {
      const v4f wq = *(const v4f*)(wp + (size_t)t * 2048);
      const v4f bq = *(const v4f*)(bp + (size_t)t * 2048);
      const v2f wlo = __builtin_shufflevector(wq, wq, 0, 1);
      const v2f whi = __builtin_shufflevector(wq, wq, 2, 3);
      const v2f blo = __builtin_shufflevector(bq, bq, 0, 1);
      const v2f bhi = __builtin_shufflevector(bq, bq, 2, 3);
      acc0[t] = __builtin_amdgcn_wmma_f32_16x16x4_f32(false, a0l, false, wlo, (short)0, acc0[t], false, false);
      acc0[t] = __builtin_amdgcn_wmma_f32_16x16x4_f32(false, x0l, false, blo, (short)0, acc0[t], false, false);
      acc1[t] = __builtin_amdgcn_wmma_f32_16x16x4_f32(false, a1l, false, wlo, (short)0, acc1[t], false, false);
      acc1[t] = __builtin_amdgcn_wmma_f32_16x16x4_f32(false, x1l, false, blo, (short)0, acc1[t], false, false);
      acc0[t] = __builtin_amdgcn_wmma_f32_16x16x4_f32(false, a0h, false, whi, (short)0, acc0[t], false, false);
      acc0[t] = __builtin_amdgcn_wmma_f32_16x16x4_f32(false, x0h, false, bhi, (short)0, acc0[t], false, false);
      acc1[t] = __builtin_amdgcn_wmma_f32_16x16x4_f32(false, a1h, false, whi, (short)0, acc1[t], false, false);
      acc1[t] = __builtin_amdgcn_wmma_f32_16x16x4_f32(false, x1h, false, bhi, (short)0, acc1[t], false, false);
    }
  }

  // D layout: VGPR r -> M = r + 8*kh ; N = (lane & 15) within tile.
  float* o0 = out + (size_t)(m0 + kh * 8) * FDIM + row;
  float* o1 = out + (size_t)(m0 + 16 + kh * 8) * FDIM + row;
#pragma unroll
  for (int t = 0; t < 8; ++t) {
#pragma unroll
    for (int r = 0; r < 8; ++r) {
      o0[(size_t)r * FDIM + t * 16] = acc0[t][r];
      o1[(size_t)r * FDIM + t * 16] = acc1[t][r];
    }
  }
}

extern "C" void kernel_launch(void* const* d_in, const int* in_sizes, int n_in,
                              void* d_out, int out_size, void* d_ws, size_t ws_size,
                              hipStream_t stream) {
  const float* x   = (const float*)d_in[0];
  const int*   ei  = (const int*)d_in[1];   // [2,E]: sources, then destinations
  const float* W   = (const float*)d_in[2];
  const float* Bm  = (const float*)d_in[3];
  float* out = (float*)d_out;

  // workspace layout
  float* agg  = (float*)d_ws;                        // N*F floats (51.2MB)
  float* wswz = agg + (size_t)NNODES * FDIM;         // 16384 floats
  float* bswz = wswz + 16384;                        // 16384 floats
  int*   cnt  = (int*)(bswz + 16384);                // N  (zeroed)
  int*   cur  = cnt + NNODES;                        // N  (zeroed; right after cnt)
  int*   offs = cur + NNODES;                        // N+1
  int*   blksum = offs + NNODES + 1;                 // NBLK
  int*   blkoff = blksum + NBLK;                     // NBLK
  int*   bind   = blkoff + NBLK;                     // E

  // 1) zero cnt + cursor (contiguous 2N ints)
  gcn_zero_kernel<<<(2 * NNODES + 255) / 256, 256, 0, stream>>>(cnt);

  // 2) swizzle W/B into WMMA operand images
  gcn_swizzle_kernel<<<32, 256, 0, stream>>>(W, Bm, wswz, bswz);

  // 3) CSR build
  const int eblocks = (NEDGES + 255) / 256;
  gcn_count_kernel<<<eblocks, 256, 0, stream>>>(ei, cnt);
  gcn_scan_reduce_kernel<<<NBLK, 256, 0, stream>>>(cnt, blksum);
  gcn_scan_top_kernel<<<1, 32, 0, stream>>>(blksum, blkoff, offs);
  gcn_scan_apply_kernel<<<NBLK, 256, 0, stream>>>(cnt, blkoff, offs);
  gcn_bin_kernel<<<eblocks, 256, 0, stream>>>(ei, offs, cur, bind);

  // 4) gather-aggregate: one wave per node, 8 waves per block
  gcn_aggregate_kernel<<<(NNODES + 7) / 8, 256, 0, stream>>>(x, offs, bind, agg);

  // 5) GEMM: one wave per 32 nodes
  const int gemm_blocks = (NNODES / 32 + 7) / 8;   // 391
  gcn_gemm_kernel<<<gemm_blocks, 256, 0, stream>>>(x, agg, wswz, bswz, out);
}